// LIF_59631325938021
// MI455X (gfx1250) — compile-verified
//
#include <hip/hip_runtime.h>

#define NN 8192

typedef float v2f __attribute__((ext_vector_type(2)));
typedef float v8f __attribute__((ext_vector_type(8)));

// LIF step: I = g @ w ; I' = sigmoid(12*I/N) + 0.9*x ; v' = v + (E_L - v + I'*(30-E_L))/tau_m ;
// out = sigmoid(v' - 30).
// GEMV done as degenerate 16x16x4 f32 WMMA: g slice in row 0 of A, 16 columns of w in B,
// accumulator row 0 = partial dot products. Bandwidth-bound on the 256MB stream of w.
__global__ __launch_bounds__(256) void lif_gemv_wmma(
    const float* __restrict__ x_in,
    const float* __restrict__ v,
    const float* __restrict__ g,
    const float* __restrict__ w,
    const float* __restrict__ E_L,
    const float* __restrict__ tau_m,
    float* __restrict__ out)
{
  __shared__ float gs[NN];        // 32 KB: whole g vector, reused by all 8 waves
  __shared__ float red[8][16];    // per-wave partial I for the block's 16 columns

  const int tid  = threadIdx.x;
  const int lane = tid & 31;
  const int wave = tid >> 5;
  const int col  = lane & 15;     // output column within tile / A-matrix M index
  const int half = lane >> 4;     // 0: lanes 0-15 (K=0,1), 1: lanes 16-31 (K=2,3)
  const int j0   = blockIdx.x * 16;

  // Stage g into LDS with wide coalesced loads.
  {
    const float4* g4  = (const float4*)g;
    float4*       gs4 = (float4*)gs;
    for (int i = tid; i < NN / 4; i += 256) gs4[i] = g4[i];
  }
  __syncthreads();

  // Each wave reduces a 1024-row K-slice of w for this 16-column tile.
  const int kBegin = wave * (NN / 8);

  // B operand layout guess mirrors the C/D half-lane split:
  //   b.x (VGPR0): lanes 0-15 -> row k+0, lanes 16-31 -> row k+2
  //   b.y (VGPR1): lanes 0-15 -> row k+1, lanes 16-31 -> row k+3
  const float* w0 = w + (size_t)(kBegin + 2 * half)     * NN + j0 + col;
  const float* w1 = w + (size_t)(kBegin + 2 * half + 1) * NN + j0 + col;

  v8f acc = {};
#pragma unroll 4
  for (int k = kBegin; k < kBegin + NN / 8; k += 4) {
    v2f b;
    b.x = *w0;                        // 64B-coalesced row segments of w
    b.y = *w1;

    // A row 0 = g[k..k+3]; all other rows zero (their outputs are ignored).
    v2f ga = *(const v2f*)&gs[k + 2 * half];   // ds_load_b64
    v2f a;
    a.x = (col == 0) ? ga.x : 0.0f;
    a.y = (col == 0) ? ga.y : 0.0f;

    acc = __builtin_amdgcn_wmma_f32_16x16x4_f32(
        /*neg_a=*/false, a, /*neg_b=*/false, b,
        /*c_mod=*/(short)0, acc, /*reuse_a=*/false, /*reuse_b=*/false);

    w0 += 4 * NN;
    w1 += 4 * NN;
  }

  // D row 0 lives in acc[0] (VGPR0), lanes 0..15 -> columns j0..j0+15.
  if (lane < 16) red[wave][lane] = acc[0];
  __syncthreads();

  if (tid < 16) {
    float I = 0.0f;
#pragma unroll
    for (int wv = 0; wv < 8; ++wv) I += red[wv][tid];

    const int j = j0 + tid;
    float Is = 1.0f / (1.0f + __expf(-(12.0f * I / (float)NN))) + 0.9f * x_in[j];
    float el = E_L[j];
    float vj = v[j];
    float vn = vj + (el - vj + Is * (30.0f - el)) / tau_m[j];
    out[j] = 1.0f / (1.0f + __expf(-(vn - 30.0f)));
  }
}

extern "C" void kernel_launch(void* const* d_in, const int* in_sizes, int n_in,
                              void* d_out, int out_size, void* d_ws, size_t ws_size,
                              hipStream_t stream) {
  // setup_inputs() order: x_in, v, g, w, E_L, tau_m, tau_g (tau_g unused by output)
  const float* x_in  = (const float*)d_in[0];
  const float* v     = (const float*)d_in[1];
  const float* g     = (const float*)d_in[2];
  const float* w     = (const float*)d_in[3];
  const float* E_L   = (const float*)d_in[4];
  const float* tau_m = (const float*)d_in[5];
  float* out = (float*)d_out;

  lif_gemv_wmma<<<NN / 16, 256, 0, stream>>>(x_in, v, g, w, E_L, tau_m, out);
}